// FermiLayer_21629455302830
// MI455X (gfx1250) — compile-verified
//
#include <hip/hip_runtime.h>
#include <hip/hip_bf16.h>

// ---------------------------------------------------------------------------
// FermiLayer for MI455X (gfx1250, wave32, WMMA).
// fp32 end-to-end: the dominant h2 pair-GEMM is ~16 FLOP/byte, at/below the
// HBM roof (23.3 TB/s), so V_WMMA_F32_16X16X4_F32 keeps reference numerics.
// Epilogue gelu uses the gfx1250 hardware tanh (branch-free) so the 37.7M
// activation evaluations don't swamp the VALU pipe.
// ---------------------------------------------------------------------------

typedef __attribute__((ext_vector_type(2))) float v2f;
typedef __attribute__((ext_vector_type(8))) float v8f;

#define NTOT    768
#define NNUC    128
#define BND2    448          // start of dn section
#define SINGLE  256
#define PAIR    64
#define DFEAT   1216
#define INV_SQRT2    0.7071067811865476f
#define GELU_RESCALE 1.7128586f

// ---- branch-free tanh: hardware V_TANH_F32 if the builtin exists ----------
// All amdgcn builtins are gated behind __HIP_DEVICE_COMPILE__: the host pass
// of hipcc does not declare them, so it gets a parse-only stub instead.
#if defined(__HIP_DEVICE_COMPILE__)
#  if __has_builtin(__builtin_amdgcn_tanhf)
__device__ __forceinline__ float fast_tanh(float x) {
    return __builtin_amdgcn_tanhf(x);
}
#  elif __has_builtin(__builtin_amdgcn_tanh_f32)
__device__ __forceinline__ float fast_tanh(float x) {
    return __builtin_amdgcn_tanh_f32(x);
}
#  else
__device__ __forceinline__ float fast_tanh(float x) {
    // tanh(x) = sign(x) * (1 - e) / (1 + e),  e = exp(-2|x|) = exp2(-2*log2e*|x|)
    float ax = __builtin_fabsf(x);
    float e  = __builtin_amdgcn_exp2f(ax * -2.8853900817779268f);
    float t  = (1.0f - e) * __builtin_amdgcn_rcpf(1.0f + e);
    return __builtin_copysignf(t, x);
}
#  endif
#else
__device__ __forceinline__ float fast_tanh(float x) { return x; }  // host stub
#endif

__device__ __forceinline__ float gelu_act(float x) {
    // jax.nn.gelu (tanh approx) * rescale, branch-free:
    // u = sqrt(2/pi) * (x + 0.044715 x^3) = x * (c0 + c2 * x^2)
    float x2 = x * x;
    float u  = x * __builtin_fmaf(x2, 0.035677408136300125f, 0.7978845608028654f);
    float t  = fast_tanh(u);
    return (0.5f * GELU_RESCALE) * x * (1.0f + t);
}

__device__ __forceinline__ v8f wmma_f32(v2f a, v2f b, v8f c) {
#if defined(__HIP_DEVICE_COMPILE__)
    // 8 args: (neg_a, A, neg_b, B, c_mod, C, reuse_a, reuse_b)
    return __builtin_amdgcn_wmma_f32_16x16x4_f32(false, a, false, b,
                                                 (short)0, c, false, false);
#else
    (void)a; (void)b;
    return c;   // host stub, never executed
#endif
}

__device__ __forceinline__ int sec_of(int i) {
    return (i < NNUC) ? 0 : ((i < BND2) ? 1 : 2);
}
__device__ __forceinline__ int pair_type(int n, int m) {
    int sn = sec_of(n), sm = sec_of(m);
    if (sn == 0 && sm == 0) return 0;       // nucl-nucl
    if (sn == 0 || sm == 0) return 1;       // nucl-elec
    return (sn == sm) ? 2 : 3;              // same-spin / opposite-spin
}

// ---------------------------------------------------------------------------
// 1) per-section mean of h1 -> h1m (3 x 256)
// ---------------------------------------------------------------------------
__global__ void k_h1mean(const float* __restrict__ h1, float* __restrict__ h1m) {
    int s = blockIdx.x, d = threadIdx.x;
    int lo = (s == 0) ? 0 : ((s == 1) ? NNUC : BND2);
    int hi = (s == 0) ? NNUC : ((s == 1) ? BND2 : NTOT);
    float acc = 0.f;
    for (int r = lo; r < hi; ++r) acc += h1[(size_t)r * SINGLE + d];
    h1m[s * SINGLE + d] = acc / (float)(hi - lo);
}

// ---------------------------------------------------------------------------
// 2) per-(row, section) mean of h2 over axis 1 -> h2m (768 x 3 x 64)
// ---------------------------------------------------------------------------
__global__ void k_h2mean(const float* __restrict__ h2, float* __restrict__ h2m) {
    int n = blockIdx.x;
    int g = threadIdx.x >> 6;       // 4 m-groups
    int d = threadIdx.x & 63;
    float s0 = 0.f, s1 = 0.f, s2 = 0.f;
    const float* base = h2 + (size_t)n * NTOT * PAIR + d;
    for (int m = g; m < NTOT; m += 4) {
        float v = base[(size_t)m * PAIR];
        if (m < NNUC)      s0 += v;
        else if (m < BND2) s1 += v;
        else               s2 += v;
    }
    __shared__ float red[4][3][64];
    red[g][0][d] = s0; red[g][1][d] = s1; red[g][2][d] = s2;
    __syncthreads();
    if (g == 0) {
        float t0 = red[0][0][d] + red[1][0][d] + red[2][0][d] + red[3][0][d];
        float t1 = red[0][1][d] + red[1][1][d] + red[2][1][d] + red[3][1][d];
        float t2 = red[0][2][d] + red[1][2][d] + red[2][2][d] + red[3][2][d];
        float* o = h2m + (size_t)n * 3 * PAIR;
        o[0 * PAIR + d] = t0 * (1.0f / 128.0f);
        o[1 * PAIR + d] = t1 * (1.0f / 320.0f);
        o[2 * PAIR + d] = t2 * (1.0f / 320.0f);
    }
}

// ---------------------------------------------------------------------------
// 3) assemble feats (768 x 1216) = [h1 | h2m(sw) | h1m(sw)], spin swap for dn
// ---------------------------------------------------------------------------
__global__ void k_feats(const float* __restrict__ h1, const float* __restrict__ h1m,
                        const float* __restrict__ h2m, float* __restrict__ feats) {
    int n = blockIdx.x, t = threadIdx.x;
    bool sw = (n >= BND2);
    float* fr = feats + (size_t)n * DFEAT;
    fr[t] = h1[(size_t)n * SINGLE + t];
    if (t < 192) {
        int j = t / 64, d = t % 64;
        int s = (j == 0) ? 0 : (sw ? (3 - j) : j);
        fr[SINGLE + t] = h2m[((size_t)n * 3 + s) * PAIR + d];
    }
    for (int idx = t; idx < 768; idx += 256) {
        int j = idx / 256, d = idx % 256;
        int s = (j == 0) ? 0 : (sw ? (3 - j) : j);
        fr[448 + idx] = h1m[s * SINGLE + d];
    }
}

// ---------------------------------------------------------------------------
// 4) h1 update: feats @ (W1n|W1e) + bias, gelu, residual.
//    One wave per 16x16 output tile, K=1216 via V_WMMA_F32_16X16X4_F32.
//    Row boundary 128 is 16-aligned -> per-tile weight selection is uniform.
// ---------------------------------------------------------------------------
__global__ void __launch_bounds__(256)
k_h1gemm(const float* __restrict__ feats,
         const float* __restrict__ W1n, const float* __restrict__ b1n,
         const float* __restrict__ W1e, const float* __restrict__ b1e,
         const float* __restrict__ h1, float* __restrict__ h1o) {
    int w    = threadIdx.x >> 5;
    int lane = threadIdx.x & 31;
    int tile = blockIdx.x * 8 + w;          // 768 tiles = 48 row x 16 col
    int rb = (tile >> 4) * 16;
    int cb = (tile & 15) * 16;
    const float* W;  const float* bias;
    if (rb < NNUC) { W = W1n; bias = b1n; } else { W = W1e; bias = b1e; }

    int M = lane & 15, kh = lane >> 4;      // A frag: row M, K = 4*ks + 2*kh + {0,1}
    const float* Arow = feats + (size_t)(rb + M) * DFEAT;
    int col = cb + M;

    v8f acc = {};
#pragma unroll 4
    for (int ks = 0; ks < DFEAT / 4; ++ks) {
        int k0 = 4 * ks + 2 * kh;
        v2f a = *(const v2f*)(Arow + k0);   // 8B-aligned (k0 even)
        v2f b;
        b.x = W[(size_t)k0 * SINGLE + col];
        b.y = W[(size_t)(k0 + 1) * SINGLE + col];
        acc = wmma_f32(a, b, acc);
    }
    float bv = bias[col];
#pragma unroll
    for (int r = 0; r < 8; ++r) {
        int row = rb + r + 8 * kh;
        float y = acc[r] + bv;
        float x = h1[(size_t)row * SINGLE + col];
        h1o[(size_t)row * SINGLE + col] = (x + gelu_act(y)) * INV_SQRT2;
    }
}

// ---------------------------------------------------------------------------
// 5) h2 update: per pair-type 64x64 GEMM on 589,824 pair rows.
//    8 waves/block, each wave = 16 pair-rows x 64 cols (64 WMMAs).
//    W2 is restriped in LDS so each lane's (k, k+1) B-pair is one b64 load,
//    with a 160-float pair-row stride (== 32 mod 64 banks) so the two
//    half-waves hit disjoint bank sets. 80 KB of the 320 KB WGP budget.
//    Section boundaries are 16-aligned -> pair type uniform per wave.
// ---------------------------------------------------------------------------
#define WROW 160   // floats per (p, k-pair) row in LDS (128 used + 32 pad)

__global__ void __launch_bounds__(256)
k_h2upd(const float* __restrict__ h2, const float* __restrict__ W2,
        const float* __restrict__ b2, float* __restrict__ h2o) {
    __shared__ float sW[4 * 32 * WROW];  // [p][kpair][2*e + (k&1)], 80 KB
    __shared__ float sB[4][64];
    for (int i = threadIdx.x; i < 4 * 64 * 64; i += 256) {
        int p = i >> 12;
        int k = (i >> 6) & 63;
        int e = i & 63;
        sW[(p * 32 + (k >> 1)) * WROW + 2 * e + (k & 1)] = W2[i];
    }
    ((float*)sB)[threadIdx.x] = b2[threadIdx.x];
    __syncthreads();

    int w    = threadIdx.x >> 5;
    int lane = threadIdx.x & 31;
    int rb = blockIdx.x * 128 + w * 16;         // flat pair-row chunk base
    int n  = rb / NTOT;
    int mb = rb % NTOT;
    int p  = pair_type(n, mb);                  // uniform over the 16 rows

    int M = lane & 15, kh = lane >> 4;
    const float* Arow = h2 + (size_t)(rb + M) * PAIR;
    const float* sWp  = sW + p * 32 * WROW;

    v8f acc0 = {}, acc1 = {}, acc2 = {}, acc3 = {};
#pragma unroll
    for (int ks = 0; ks < 16; ++ks) {
        int k0 = 4 * ks + 2 * kh;
        v2f a = *(const v2f*)(Arow + k0);       // each h2 elem read once/wave
        const float* wrow = sWp + (2 * ks + kh) * WROW;   // k-pair row
        v2f b0  = *(const v2f*)(wrow + 2 * ( 0 + M));
        v2f b1  = *(const v2f*)(wrow + 2 * (16 + M));
        v2f b2v = *(const v2f*)(wrow + 2 * (32 + M));
        v2f b3  = *(const v2f*)(wrow + 2 * (48 + M));
        acc0 = wmma_f32(a, b0,  acc0);
        acc1 = wmma_f32(a, b1,  acc1);
        acc2 = wmma_f32(a, b2v, acc2);
        acc3 = wmma_f32(a, b3,  acc3);
    }

#pragma unroll
    for (int r = 0; r < 8; ++r) {
        int row = rb + r + 8 * kh;              // C/D layout: VGPR r -> M=r(+8)
        const float* hrow = h2  + (size_t)row * PAIR;   // L0-hot reload
        float*       orow = h2o + (size_t)row * PAIR;
        float y0 = acc0[r] + sB[p][ 0 + M];
        float y1 = acc1[r] + sB[p][16 + M];
        float y2 = acc2[r] + sB[p][32 + M];
        float y3 = acc3[r] + sB[p][48 + M];
        orow[ 0 + M] = (hrow[ 0 + M] + gelu_act(y0)) * INV_SQRT2;
        orow[16 + M] = (hrow[16 + M] + gelu_act(y1)) * INV_SQRT2;
        orow[32 + M] = (hrow[32 + M] + gelu_act(y2)) * INV_SQRT2;
        orow[48 + M] = (hrow[48 + M] + gelu_act(y3)) * INV_SQRT2;
    }
}

// ---------------------------------------------------------------------------
extern "C" void kernel_launch(void* const* d_in, const int* in_sizes, int n_in,
                              void* d_out, int out_size, void* d_ws, size_t ws_size,
                              hipStream_t stream) {
    const float* h1  = (const float*)d_in[0];
    const float* h2  = (const float*)d_in[1];
    const float* W1n = (const float*)d_in[2];
    const float* b1n = (const float*)d_in[3];
    const float* W1e = (const float*)d_in[4];
    const float* b1e = (const float*)d_in[5];
    const float* W2  = (const float*)d_in[6];
    const float* b2  = (const float*)d_in[7];

    float* h1_out = (float*)d_out;
    float* h2_out = (float*)d_out + (size_t)NTOT * SINGLE;

    // workspace layout (floats): h1m[3*256] | h2m[768*3*64] | feats[768*1216]
    float* ws    = (float*)d_ws;
    float* h1m   = ws;
    float* h2m   = ws + 3 * SINGLE;
    float* feats = h2m + (size_t)NTOT * 3 * PAIR;

    k_h1mean<<<3,   256, 0, stream>>>(h1, h1m);
    k_h2mean<<<NTOT,256, 0, stream>>>(h2, h2m);
    k_feats <<<NTOT,256, 0, stream>>>(h1, h1m, h2m, feats);
    k_h1gemm<<<96,  256, 0, stream>>>(feats, W1n, b1n, W1e, b1e, h1, h1_out);
    k_h2upd <<<(NTOT * NTOT) / 128, 256, 0, stream>>>(h2, W2, b2, h2_out);
}